// RMSELoss_39273180954721
// MI455X (gfx1250) — compile-verified
//
#include <hip/hip_runtime.h>
#include <math.h>

// ---------------------------------------------------------------------------
// RMSE of per-(batch, plot-id) segmented sums.
//   B=32, H=W=1024, NPIDS=128  (fixed by the reference harness)
//   ws layout (floats): [0,4096)=sum(yhat), [4096,8192)=sum(y), [8192,12288)=count
// ---------------------------------------------------------------------------

#define BATCH     32
#define NPIDS     128
#define NSEG      (BATCH * NPIDS)          // 4096
#define PIX_ROW   (1024 * 1024)            // pixels per batch row
#define SPLIT     32                       // blocks per batch row
#define PIX_BLK   (PIX_ROW / SPLIT)        // 32768 pixels per block
#define IGNORE_I  (-999)
#define IGNORE_F  (-999.0f)
#define EPS_F     (1e-6f)

typedef __attribute__((ext_vector_type(2))) float v2f;
typedef __attribute__((ext_vector_type(4))) float v4f;
typedef __attribute__((ext_vector_type(8))) float v8f;
typedef __attribute__((ext_vector_type(4))) int   v4i;

__global__ __launch_bounds__(256) void rmse_zero_ws(float* __restrict__ ws, int n) {
    int i = blockIdx.x * blockDim.x + threadIdx.x;
    if (i < n) ws[i] = 0.0f;
}

__device__ __forceinline__ void px_accum(float yp, float yt, int p,
                                         float* h_sp, float* h_st, float* h_ct) {
    if (p != IGNORE_I) {
        atomicAdd(&h_sp[p], yp);   // ds_add_f32 (non-returning)
        atomicAdd(&h_st[p], yt);
        atomicAdd(&h_ct[p], 1.0f);
    }
}

__global__ __launch_bounds__(256) void rmse_accum(const float* __restrict__ yhat,
                                                  const float* __restrict__ y,
                                                  const int*   __restrict__ pid,
                                                  float* __restrict__ ws) {
    __shared__ float h_sp[NPIDS];
    __shared__ float h_st[NPIDS];
    __shared__ float h_ct[NPIDS];

    const int t = threadIdx.x;
    for (int i = t; i < NPIDS; i += 256) { h_sp[i] = 0.f; h_st[i] = 0.f; h_ct[i] = 0.f; }
    __syncthreads();

    const int b = blockIdx.x / SPLIT;      // batch row this block is pinned to
    const int c = blockIdx.x % SPLIT;      // chunk within the row
    // float4-granular base index (everything is 16B aligned by construction)
    const int base4 = b * (PIX_ROW / 4) + c * (PIX_BLK / 4);

    const v4f* __restrict__ yp4 = (const v4f*)yhat;
    const v4f* __restrict__ yt4 = (const v4f*)y;
    const v4i* __restrict__ pm4 = (const v4i*)pid;

    // 32 iterations x 256 threads x 4 floats = 32768 pixels, fully coalesced.
    // Non-temporal: 402MB single-pass stream, keep it out of L2 residency.
    for (int k = 0; k < PIX_BLK / (4 * 256); ++k) {
        const int i4 = base4 + k * 256 + t;
        v4f a  = __builtin_nontemporal_load(&yp4[i4]);   // global_load_b128, TH=NT
        v4f bb = __builtin_nontemporal_load(&yt4[i4]);
        v4i p  = __builtin_nontemporal_load(&pm4[i4]);
        px_accum(a[0], bb[0], p[0], h_sp, h_st, h_ct);
        px_accum(a[1], bb[1], p[1], h_sp, h_st, h_ct);
        px_accum(a[2], bb[2], p[2], h_sp, h_st, h_ct);
        px_accum(a[3], bb[3], p[3], h_sp, h_st, h_ct);
    }

    __syncthreads();
    // Flush this block's 128-bin histogram into the global segment arrays.
    for (int i = t; i < NPIDS; i += 256) {
        float cv = h_ct[i];
        if (cv != 0.0f) {
            const int s = b * NPIDS + i;
            atomicAdd(&ws[s],            h_sp[i]);   // global_atomic_add_f32
            atomicAdd(&ws[NSEG + s],     h_st[i]);
            atomicAdd(&ws[2 * NSEG + s], h_ct[i]);
        }
    }
}

// Single wave (32 threads). Per-lane partials over 4096 segments, then one
// V_WMMA_F32_16X16X4_F32 reduces both partial vectors at once:
//   A[m][k]: lanes 0-15 hold {K0,K1}, lanes 16-31 hold {K2,K3} (ISA layout),
//   per-lane a = {sq_partial (even K), keep_partial (odd K)}.
//   B[k][n]: column 0 = ones on even K rows, column 1 = ones on odd K rows
//   => D[m][0] = sq(m)+sq(m+16), D[m][1] = keep(m)+keep(m+16).
__global__ __launch_bounds__(32) void rmse_final(const float* __restrict__ ws,
                                                 float* __restrict__ out) {
    const int lane = threadIdx.x;

    float sq = 0.0f, kp = 0.0f;
    #pragma unroll 4
    for (int s = lane; s < NSEG; s += 32) {
        float sp = ws[s];
        float st = ws[NSEG + s];
        float ct = ws[2 * NSEG + s];
        bool keep = (ct > 0.0f) && (st > 0.0f) && (st != IGNORE_F);
        float d = sp - st;
        sq += keep ? d * d : 0.0f;
        kp += keep ? 1.0f : 0.0f;
    }

    v2f a; a[0] = sq; a[1] = kp;
    const int n = lane & 15;
    v2f bm; bm[0] = (n == 0) ? 1.0f : 0.0f;   // even-K row of B -> column 0
            bm[1] = (n == 1) ? 1.0f : 0.0f;   // odd-K  row of B -> column 1
    v8f c = {};
    v8f d = __builtin_amdgcn_wmma_f32_16x16x4_f32(
        /*neg_a=*/false, a, /*neg_b=*/false, bm,
        /*c_mod=*/(short)0, c, /*reuse_a=*/false, /*reuse_b=*/false);

    // Each lane sums its 8 C/D rows; col0 lives in lanes {0,16}, col1 in {1,17}.
    float s = 0.0f;
    #pragma unroll
    for (int i = 0; i < 8; ++i) s += d[i];

    float s16 = __shfl(s, 16, 32);
    float s1  = __shfl(s, 1, 32);
    float s17 = __shfl(s, 17, 32);
    if (lane == 0) {
        float sq_tot = s + s16;
        float kp_tot = s1 + s17;
        float nkeep  = fmaxf(kp_tot, 1.0f);
        out[0] = sqrtf(sq_tot / nkeep + EPS_F);
    }
}

extern "C" void kernel_launch(void* const* d_in, const int* in_sizes, int n_in,
                              void* d_out, int out_size, void* d_ws, size_t ws_size,
                              hipStream_t stream) {
    const float* yhat = (const float*)d_in[0];
    const float* y    = (const float*)d_in[1];
    const int*   pm   = (const int*)d_in[2];
    float* ws  = (float*)d_ws;
    float* out = (float*)d_out;

    const int nws = 3 * NSEG;  // 12288 floats of scratch
    rmse_zero_ws<<<(nws + 255) / 256, 256, 0, stream>>>(ws, nws);
    rmse_accum<<<BATCH * SPLIT, 256, 0, stream>>>(yhat, y, pm, ws);
    rmse_final<<<1, 32, 0, stream>>>(ws, out);
}